// EdgeGCNetwork_51393578664471
// MI455X (gfx1250) — compile-verified
//
#include <hip/hip_runtime.h>

typedef float v2f __attribute__((ext_vector_type(2)));
typedef float v8f __attribute__((ext_vector_type(8)));

#define NN 100000
#define NE 1600000
#define D  64

// ---------------------------------------------------------------------------
// P[n,:] = (X[n,:] * norm[n]) @ W    (W is 64x64, staged in LDS)
// One wave -> 16x64 output tile via V_WMMA_F32_16X16X4_F32.
// A-pairs for all 16 K-steps preloaded into registers (no per-step stalls).
// ---------------------------------------------------------------------------
__global__ __launch_bounds__(256) void gcn_proj(const float* __restrict__ X,
                                                const float* __restrict__ norm,
                                                const float* __restrict__ W,
                                                float* __restrict__ P)
{
    __shared__ float Wl[D * D];   // 16 KB
    const int t = threadIdx.x;

    // cooperative load of W into LDS: 4096 floats / 256 threads = 4 x float4
    {
        const float4* Wg = (const float4*)W;
        float4* Ws = (float4*)Wl;
#pragma unroll
        for (int i = 0; i < 4; ++i)
            Ws[t + 256 * i] = Wg[t + 256 * i];
    }
    __syncthreads();

    const int wave = t >> 5;            // 0..7
    const int lane = t & 31;
    const int l15  = lane & 15;
    const int hi   = lane >> 4;         // 0: lanes 0-15, 1: lanes 16-31
    const long rowBase = (long)blockIdx.x * 128 + (long)wave * 16;

    // A-fragment source row for this lane (16x4 f32 A layout: both half-waves
    // cover M=0..15; hi half holds K=2,3).  Clamp row so loads are always
    // in-bounds; invalid rows contribute 0 via nrm = 0.
    const long r      = rowBase + l15;
    const bool rvalid = (r < NN);
    const long rr     = rvalid ? r : (NN - 1);
    const float nrm   = rvalid ? norm[r] : 0.0f;

    // Preload all 16 A-pairs: k-step k needs floats [4k+2hi, 4k+2hi+1]
    const float2* xp = (const float2*)(X + rr * D);
    v2f areg[16];
#pragma unroll
    for (int k = 0; k < 16; ++k) {
        const float2 v = xp[2 * k + hi];
        areg[k].x = v.x * nrm;
        areg[k].y = v.y * nrm;
    }

    v8f c0 = {}, c1 = {}, c2 = {}, c3 = {};

#pragma unroll
    for (int k = 0; k < 16; ++k) {
        const int kk = 4 * k + hi * 2;
        v2f b;
        b.x = Wl[(kk + 0) * D +  0 + l15];
        b.y = Wl[(kk + 1) * D +  0 + l15];
        c0 = __builtin_amdgcn_wmma_f32_16x16x4_f32(false, areg[k], false, b, (short)0, c0, false, false);
        b.x = Wl[(kk + 0) * D + 16 + l15];
        b.y = Wl[(kk + 1) * D + 16 + l15];
        c1 = __builtin_amdgcn_wmma_f32_16x16x4_f32(false, areg[k], false, b, (short)0, c1, false, false);
        b.x = Wl[(kk + 0) * D + 32 + l15];
        b.y = Wl[(kk + 1) * D + 32 + l15];
        c2 = __builtin_amdgcn_wmma_f32_16x16x4_f32(false, areg[k], false, b, (short)0, c2, false, false);
        b.x = Wl[(kk + 0) * D + 48 + l15];
        b.y = Wl[(kk + 1) * D + 48 + l15];
        c3 = __builtin_amdgcn_wmma_f32_16x16x4_f32(false, areg[k], false, b, (short)0, c3, false, false);
    }

    // C/D layout: VGPR v -> M = v (lanes 0-15) or v+8 (lanes 16-31), N = l15
    if (rowBase + 16 <= NN) {
#pragma unroll
        for (int v = 0; v < 8; ++v) {
            float* prow = P + (rowBase + v + hi * 8) * D + l15;
            prow[ 0] = c0[v];
            prow[16] = c1[v];
            prow[32] = c2[v];
            prow[48] = c3[v];
        }
    } else {
#pragma unroll
        for (int v = 0; v < 8; ++v) {
            const long m = rowBase + v + hi * 8;
            if (m < NN) {
                float* prow = P + m * D + l15;
                prow[ 0] = c0[v];
                prow[16] = c1[v];
                prow[32] = c2[v];
                prow[48] = c3[v];
            }
        }
    }
}

// ---------------------------------------------------------------------------
// CSR build (once per call; graph reused by all 3 layers)
// ---------------------------------------------------------------------------
__global__ __launch_bounds__(256) void csr_hist(const int* __restrict__ dst,
                                                int* __restrict__ off)
{
    const long e = (long)blockIdx.x * 256 + threadIdx.x;
    if (e < NE) atomicAdd(&off[dst[e] + 1], 1);
}

// Single-block inclusive scan of off[0..n-1] (off[0]=0, off[i]=count[i-1]
// before the scan -> CSR row offsets after the scan).
__global__ __launch_bounds__(1024) void csr_scan(int* __restrict__ off, int n)
{
    __shared__ int tmp[1024];
    __shared__ int carry;
    const int t = threadIdx.x;
    if (t == 0) carry = 0;
    __syncthreads();
    for (int base = 0; base < n; base += 1024) {
        const int i = base + t;
        int v = (i < n) ? off[i] : 0;
        tmp[t] = v;
        __syncthreads();
#pragma unroll
        for (int s = 1; s < 1024; s <<= 1) {
            const int x = (t >= s) ? tmp[t - s] : 0;
            __syncthreads();
            tmp[t] += x;
            __syncthreads();
        }
        const int res = tmp[t] + carry;
        if (i < n) off[i] = res;
        __syncthreads();
        if (t == 1023) carry = res;
        __syncthreads();
    }
}

__global__ __launch_bounds__(256) void csr_fill(const int* __restrict__ src,
                                                const int* __restrict__ dst,
                                                int* __restrict__ cursor,
                                                int* __restrict__ sorted_src)
{
    const long e = (long)blockIdx.x * 256 + threadIdx.x;
    if (e < NE) {
        const int d   = dst[e];
        const int pos = atomicAdd(&cursor[d], 1);
        sorted_src[pos] = src[e];
    }
}

// ---------------------------------------------------------------------------
// Y[n,:] = relu?( norm[n] * sum_{e: dst==n} P[src_e,:] + b )
// 16 lanes per node, float4 strip per lane; zero float atomics.
// ---------------------------------------------------------------------------
__global__ __launch_bounds__(256) void gcn_agg(const float* __restrict__ P,
                                               const int* __restrict__ off,
                                               const int* __restrict__ ssrc,
                                               const float* __restrict__ norm,
                                               const float* __restrict__ bias,
                                               float* __restrict__ Y,
                                               int relu)
{
    const long gtid = (long)blockIdx.x * 256 + threadIdx.x;
    const long n = gtid >> 4;
    const int  c = (int)(gtid & 15);
    if (n >= NN) return;

    const int beg = off[n];
    const int end = off[n + 1];
    float4 acc = make_float4(0.f, 0.f, 0.f, 0.f);
    for (int i = beg; i < end; ++i) {
        const int s = ssrc[i];
        const float4 m = *(const float4*)(P + (long)s * D + c * 4);
        acc.x += m.x; acc.y += m.y; acc.z += m.z; acc.w += m.w;
    }

    const float nrm = norm[n];
    const float4 bb = *(const float4*)(bias + c * 4);
    float4 y;
    y.x = acc.x * nrm + bb.x;
    y.y = acc.y * nrm + bb.y;
    y.z = acc.z * nrm + bb.z;
    y.w = acc.w * nrm + bb.w;
    if (relu) {
        y.x = y.x > 0.f ? y.x : 0.f;
        y.y = y.y > 0.f ? y.y : 0.f;
        y.z = y.z > 0.f ? y.z : 0.f;
        y.w = y.w > 0.f ? y.w : 0.f;
    }
    *(float4*)(Y + n * D + c * 4) = y;
}

// ---------------------------------------------------------------------------
extern "C" void kernel_launch(void* const* d_in, const int* in_sizes, int n_in,
                              void* d_out, int out_size, void* d_ws, size_t ws_size,
                              hipStream_t stream)
{
    const float* feat = (const float*)d_in[0];
    const float* nrm  = (const float*)d_in[1];
    const int*   src  = (const int*)d_in[2];
    const int*   dst  = (const int*)d_in[3];
    const float* W0   = (const float*)d_in[4];
    const float* b0   = (const float*)d_in[5];
    const float* W1   = (const float*)d_in[6];
    const float* b1   = (const float*)d_in[7];
    const float* W2   = (const float*)d_in[8];
    const float* b2   = (const float*)d_in[9];
    float* out = (float*)d_out;

    // workspace layout
    const size_t featElems = (size_t)NN * D;
    float* B0 = (float*)d_ws;                       // 25.6 MB
    float* B1 = B0 + featElems;                     // 25.6 MB
    int*   off        = (int*)(B1 + featElems);     // NN+1 ints
    int*   cursor     = off + (NN + 1);             // NN ints
    int*   sorted_src = cursor + NN;                // NE ints

    const dim3 blk(256);
    const dim3 gProj((NN + 127) / 128);
    const dim3 gEdge((NE + 255) / 256);
    const dim3 gNode(((long)NN * 16 + 255) / 256);

    // ---- CSR build (once; graph shared by all layers) ----
    hipMemsetAsync(off, 0, (NN + 1) * sizeof(int), stream);
    csr_hist<<<gEdge, blk, 0, stream>>>(dst, off);
    csr_scan<<<dim3(1), dim3(1024), 0, stream>>>(off, NN + 1);
    hipMemcpyAsync(cursor, off, NN * sizeof(int), hipMemcpyDeviceToDevice, stream);
    csr_fill<<<gEdge, blk, 0, stream>>>(src, dst, cursor, sorted_src);

    // ---- Layer 0: feat -> P=B0 -> H1=B1 (relu) ----
    gcn_proj<<<gProj, blk, 0, stream>>>(feat, nrm, W0, B0);
    gcn_agg<<<gNode, blk, 0, stream>>>(B0, off, sorted_src, nrm, b0, B1, 1);

    // ---- Layer 1: B1 -> P=B0 -> H2=B1 (relu; proj done before agg writes) ----
    gcn_proj<<<gProj, blk, 0, stream>>>(B1, nrm, W1, B0);
    gcn_agg<<<gNode, blk, 0, stream>>>(B0, off, sorted_src, nrm, b1, B1, 1);

    // ---- Layer 2: B1 -> P=B0 -> out (no relu) ----
    gcn_proj<<<gProj, blk, 0, stream>>>(B1, nrm, W2, B0);
    gcn_agg<<<gNode, blk, 0, stream>>>(B0, off, sorted_src, nrm, b2, out, 0);
}